// ActionBackgroundExamplarContrastiveLoss_68856915690199
// MI455X (gfx1250) — compile-verified
//
#include <hip/hip_runtime.h>
#include <hip/hip_bf16.h>
#include <math.h>

#define NUM_EX 200
#define DIM    128
#define NUM_F  200000
#define TILES  13          // ceil(200/16)
#define MPAD   (TILES*16)  // 208

typedef __attribute__((ext_vector_type(2))) float v2f;
typedef __attribute__((ext_vector_type(8))) float v8f;

// ---------------- zero the workspace accumulators ----------------
__global__ void zero_ws_kernel(float* __restrict__ ws, int n) {
    int i = blockIdx.x * blockDim.x + threadIdx.x;
    if (i < n) ws[i] = 0.0f;
}

// ---------------- pass 1: bandwidth-bound segment sum ----------------
// Each block accumulates into a private LDS histogram [200][128] with
// ds_add_f32, then flushes once with global_atomic_add_f32.
__global__ __launch_bounds__(256) void segsum_kernel(
    const float* __restrict__ feats, const int* __restrict__ idxs,
    float* __restrict__ gSums, float* __restrict__ gCnt) {
    __shared__ float sAcc[NUM_EX * DIM];   // 102400 B (CDNA5: 320 KB/WGP)
    __shared__ float sCnt[NUM_EX];

    for (int i = threadIdx.x; i < NUM_EX * DIM; i += 256) sAcc[i] = 0.0f;
    if (threadIdx.x < NUM_EX) sCnt[threadIdx.x] = 0.0f;
    __syncthreads();

    const int wave = threadIdx.x >> 5;
    const int lane = threadIdx.x & 31;
    const int stride = gridDim.x * 8;

    for (int row = blockIdx.x * 8 + wave; row < NUM_F; row += stride) {
        const int c = idxs[row];                          // wave-uniform
        const float4 v = ((const float4*)(feats + (size_t)row * DIM))[lane];
        float* dst = &sAcc[c * DIM + lane * 4];           // 32 distinct banks
        atomicAdd(dst + 0, v.x);
        atomicAdd(dst + 1, v.y);
        atomicAdd(dst + 2, v.z);
        atomicAdd(dst + 3, v.w);
        if (lane == 0) atomicAdd(&sCnt[c], 1.0f);
    }
    __syncthreads();

    for (int i = threadIdx.x; i < NUM_EX * DIM; i += 256)
        unsafeAtomicAdd(&gSums[i], sAcc[i]);
    if (threadIdx.x < NUM_EX)
        unsafeAtomicAdd(&gCnt[threadIdx.x], sCnt[threadIdx.x]);
}

// ---------------- fp32 WMMA tile: D = means_tile @ exN_tile^T ----------------
// A (16x4 f32): lane l holds M = l%16; VGPR r holds K = 4*ks + r + 2*(l/16).
// B (4x16 f32): lane l holds N = l%16; VGPR r holds K = 4*ks + r + 2*(l/16).
// D (16x16 f32): VGPR r, lane l -> M = r + 8*(l/16), N = l%16.
__device__ inline v8f compute_logits_tile(
    const float* __restrict__ sums, const float* __restrict__ ex,
    const float* sInvCnt, const float* sInvNorm,
    int ti, int tj, int l16, int half) {
    const int m = ti * 16 + l16;
    const int n = tj * 16 + l16;
    const int mc = (m < NUM_EX) ? m : (NUM_EX - 1);
    const int nc = (n < NUM_EX) ? n : (NUM_EX - 1);
    const float icA = sInvCnt[mc];
    const float inB = sInvNorm[nc];
    const bool mOK = (m < NUM_EX);
    const bool nOK = (n < NUM_EX);

    v8f acc = {};
    for (int ks = 0; ks < DIM / 4; ++ks) {
        const int k = ks * 4 + 2 * half;
        v2f a, b;
        a[0] = sums[mc * DIM + k]     * icA;
        a[1] = sums[mc * DIM + k + 1] * icA;
        b[0] = ex[nc * DIM + k]       * inB;
        b[1] = ex[nc * DIM + k + 1]   * inB;
        if (!mOK) { a[0] = 0.0f; a[1] = 0.0f; }
        if (!nOK) { b[0] = 0.0f; b[1] = 0.0f; }
        acc = __builtin_amdgcn_wmma_f32_16x16x4_f32(
            /*neg_a=*/false, a, /*neg_b=*/false, b,
            /*c_mod=*/(short)0, acc, /*reuse_a=*/false, /*reuse_b=*/false);
    }
    return acc;
}

// ---------------- pass 2: GEMM (WMMA) + log-softmax sum, one block ----------------
__global__ __launch_bounds__(256) void loss_kernel(
    const float* __restrict__ sums, const float* __restrict__ cnts,
    const float* __restrict__ ex, float* __restrict__ out) {
    __shared__ float sInvCnt[NUM_EX];
    __shared__ float sInvNorm[NUM_EX];
    __shared__ float sPartA[MPAD * TILES];   // per-tile row max, reused for exp sums
    __shared__ float sPartB[MPAD * TILES];   // per-tile row sums
    __shared__ float sRowMax[MPAD];
    __shared__ float sRowSum[MPAD];
    __shared__ float sRed[256];

    const int t    = threadIdx.x;
    const int wave = t >> 5;
    const int lane = t & 31;
    const int half = lane >> 4;
    const int l16  = lane & 15;
    const float invTemp = 1.0f / 0.07f;

    if (t < NUM_EX) sInvCnt[t] = 1.0f / cnts[t];
    // examplar row 1/norms (one wave per row, 32 lanes x float4 = 128 elems)
    for (int r = wave; r < NUM_EX; r += 8) {
        const float4 v = ((const float4*)(ex + (size_t)r * DIM))[lane];
        float s = v.x * v.x + v.y * v.y + v.z * v.z + v.w * v.w;
        for (int off = 1; off < 32; off <<= 1) s += __shfl_xor(s, off, 32);
        if (lane == 0) sInvNorm[r] = 1.0f / fmaxf(sqrtf(s), 1e-12f);
    }
    __syncthreads();

    // Phase 1: per-tile row max + row sum of logits
    for (int tile = wave; tile < TILES * TILES; tile += 8) {
        const int ti = tile / TILES, tj = tile % TILES;
        v8f acc = compute_logits_tile(sums, ex, sInvCnt, sInvNorm, ti, tj, l16, half);
        const bool colOK = (tj * 16 + l16) < NUM_EX;
        for (int r = 0; r < 8; ++r) {
            const float val = acc[r] * invTemp;
            float mx = colOK ? val : -3.4e38f;
            float sm = colOK ? val : 0.0f;
            for (int off = 1; off < 16; off <<= 1) {       // reduce inside each half
                mx = fmaxf(mx, __shfl_xor(mx, off, 32));
                sm += __shfl_xor(sm, off, 32);
            }
            if (l16 == 0) {
                const int rowm = ti * 16 + r + 8 * half;
                sPartA[rowm * TILES + tj] = mx;
                sPartB[rowm * TILES + tj] = sm;
            }
        }
    }
    __syncthreads();
    if (t < NUM_EX) {
        float mx = -3.4e38f, sm = 0.0f;
        for (int j = 0; j < TILES; ++j) {
            mx = fmaxf(mx, sPartA[t * TILES + j]);
            sm += sPartB[t * TILES + j];
        }
        sRowMax[t] = mx;
        sRowSum[t] = sm;
    }
    __syncthreads();

    // Phase 2: recompute tiles, accumulate exp(logit - rowMax)
    for (int tile = wave; tile < TILES * TILES; tile += 8) {
        const int ti = tile / TILES, tj = tile % TILES;
        v8f acc = compute_logits_tile(sums, ex, sInvCnt, sInvNorm, ti, tj, l16, half);
        const bool colOK = (tj * 16 + l16) < NUM_EX;
        for (int r = 0; r < 8; ++r) {
            const int rowm = ti * 16 + r + 8 * half;
            const bool rowOK = rowm < NUM_EX;
            const float rm = sRowMax[rowOK ? rowm : 0];
            const float val = acc[r] * invTemp;
            float e = (colOK && rowOK) ? expf(val - rm) : 0.0f;
            for (int off = 1; off < 16; off <<= 1) e += __shfl_xor(e, off, 32);
            if (l16 == 0) sPartA[rowm * TILES + tj] = e;   // reuse as exp sums
        }
    }
    __syncthreads();

    // Phase 3: per-row contribution, then block reduction to the scalar
    if (t < NUM_EX) {
        float es = 0.0f;
        for (int j = 0; j < TILES; ++j) es += sPartA[t * TILES + j];
        sRed[t] = sRowSum[t] - (float)NUM_EX * sRowMax[t] - (float)NUM_EX * logf(es);
    } else {
        sRed[t] = 0.0f;
    }
    __syncthreads();
    for (int s = 128; s > 0; s >>= 1) {
        if (t < s) sRed[t] += sRed[t + s];
        __syncthreads();
    }
    if (t == 0)
        out[0] = -(0.07f / 0.07f) * sRed[0] / (float)(NUM_EX * NUM_EX);
}

extern "C" void kernel_launch(void* const* d_in, const int* in_sizes, int n_in,
                              void* d_out, int out_size, void* d_ws, size_t ws_size,
                              hipStream_t stream) {
    const float* feats = (const float*)d_in[0];
    const int*   idxs  = (const int*)d_in[1];
    const float* ex    = (const float*)d_in[2];
    float* ws    = (float*)d_ws;
    float* gSums = ws;                   // 25600 floats
    float* gCnt  = ws + NUM_EX * DIM;    // 200 floats
    const int nzero = NUM_EX * DIM + NUM_EX;

    zero_ws_kernel<<<(nzero + 255) / 256, 256, 0, stream>>>(ws, nzero);
    segsum_kernel<<<256, 256, 0, stream>>>(feats, idxs, gSums, gCnt);
    loss_kernel<<<1, 256, 0, stream>>>(gSums, gCnt, ex, (float*)d_out);
}